// NeoGNNLayer_66992899883194
// MI455X (gfx1250) — compile-verified
//
#include <hip/hip_runtime.h>
#include <hip/hip_bf16.h>

// CDNA5 / gfx1250 fused GNN layer (GCN + SAGE + GIN + GAT), fp32 throughout.
// Dense GEMMs use V_WMMA_F32_16X16X4_F32 (exact fp32 matrix path, wave32).
// Sparse phases use global f32 atomics; all hot buffers are L2-resident
// (192 MB L2 >> ~132 MB workspace, weights 64 KB).

typedef __attribute__((ext_vector_type(2))) float v2f;
typedef __attribute__((ext_vector_type(8))) float v8f;

#define DF 128

// ---------------- utility kernels ----------------

__global__ void k_zero(float* __restrict__ p, size_t n) {
  size_t i = (size_t)blockIdx.x * blockDim.x + threadIdx.x;
  size_t s = (size_t)gridDim.x * blockDim.x;
  for (; i < n; i += s) p[i] = 0.f;
}

__global__ void k_cnt(const int* __restrict__ col, float* __restrict__ cnt, int E) {
  int e = blockIdx.x * blockDim.x + threadIdx.x;
  if (e < E) atomicAdd(&cnt[col[e]], 1.f);
}

// GCN deg = indeg + 1 (self loop); SAGE mean divisor = max(indeg, 1)
__global__ void k_dinv(const float* __restrict__ cnt, float* __restrict__ dinv,
                       float* __restrict__ rinv, int N) {
  int i = blockIdx.x * blockDim.x + threadIdx.x;
  if (i < N) {
    float c = cnt[i];
    dinv[i] = rsqrtf(c + 1.f);
    rinv[i] = 1.f / fmaxf(c, 1.f);
  }
}

// ---------------- WMMA fp32 GEMM: Out[16x128 strip] = A(N x 128) @ W(128 x 128) ----------------
// AMODE 0: A = A0 ; AMODE 1: A = A0 + A1 (GIN x+agg) ; AMODE 2: A = A0 * rowscale[row] (SAGE mean)
// ACCUM: load C from Out (acc += ...). BIASRELU: epilogue relu(x + bias) (GIN layer 1).
template <int AMODE, bool ACCUM, bool BIASRELU>
__global__ void k_gemm(const float* __restrict__ A0, const float* __restrict__ A1,
                       const float* __restrict__ rowscale,
                       const float* __restrict__ W, const float* __restrict__ bias,
                       float* __restrict__ Out, int nrows) {
  const int wave = threadIdx.x >> 5;
  const int lane = threadIdx.x & 31;
  const int tile = blockIdx.x * (blockDim.x >> 5) + wave;
  const int row0 = tile * 16;
  if (row0 >= nrows) return;          // N = 50000 is a multiple of 16: no ragged tiles
  const int half = lane >> 4;         // 0: lanes 0-15, 1: lanes 16-31
  const int l16  = lane & 15;

  v8f c[8];
#pragma unroll
  for (int j = 0; j < 8; ++j) {
    if (ACCUM) {
#pragma unroll
      for (int v = 0; v < 8; ++v)
        c[j][v] = Out[(size_t)(row0 + v + 8 * half) * DF + j * 16 + l16];
    } else {
      v8f z = {0.f, 0.f, 0.f, 0.f, 0.f, 0.f, 0.f, 0.f};
      c[j] = z;
    }
  }

  const int am = row0 + l16;                    // A row held by this lane
  const float* ar0 = A0 + (size_t)am * DF;
  const float* ar1 = (AMODE == 1) ? (A1 + (size_t)am * DF) : nullptr;
  const float sc = (AMODE == 2) ? rowscale[am] : 1.f;

#pragma unroll 4
  for (int ks = 0; ks < 32; ++ks) {            // K = 128 in steps of 4
    const int kb = ks * 4 + half * 2;          // lanes 0-15: K0,K1 ; lanes 16-31: K2,K3
    v2f a;
    if (AMODE == 0)      { a.x = ar0[kb];            a.y = ar0[kb + 1]; }
    else if (AMODE == 1) { a.x = ar0[kb] + ar1[kb];  a.y = ar0[kb + 1] + ar1[kb + 1]; }
    else                 { a.x = ar0[kb] * sc;       a.y = ar0[kb + 1] * sc; }
#pragma unroll
    for (int j = 0; j < 8; ++j) {              // 8 column tiles, A fragment reused
      v2f b;
      b.x = W[(size_t)kb * DF + j * 16 + l16];
      b.y = W[(size_t)(kb + 1) * DF + j * 16 + l16];
      c[j] = __builtin_amdgcn_wmma_f32_16x16x4_f32(
          false, a, false, b, (short)0, c[j], false, false);
    }
  }

#pragma unroll
  for (int j = 0; j < 8; ++j)
#pragma unroll
    for (int v = 0; v < 8; ++v) {
      float val = c[j][v];
      if (BIASRELU) { val += bias[j * 16 + l16]; val = fmaxf(val, 0.f); }
      Out[(size_t)(row0 + v + 8 * half) * DF + j * 16 + l16] = val;
    }
}

// ---------------- GAT attention dots: a_src[i]=hg[i].att_s, a_dst[i]=hg[i].att_d ----------------
__global__ void k_att(const float* __restrict__ hg, const float* __restrict__ att_s,
                      const float* __restrict__ att_d, float* __restrict__ asrc,
                      float* __restrict__ adst, int N) {
  int wave = threadIdx.x >> 5, lane = threadIdx.x & 31;
  int i = blockIdx.x * (blockDim.x >> 5) + wave;
  if (i >= N) return;
  float4 h = ((const float4*)(hg + (size_t)i * DF))[lane];
  float4 s = ((const float4*)att_s)[lane];
  float4 d = ((const float4*)att_d)[lane];
  float ps = h.x * s.x + h.y * s.y + h.z * s.z + h.w * s.w;
  float pd = h.x * d.x + h.y * d.y + h.z * d.z + h.w * d.w;
#pragma unroll
  for (int off = 16; off; off >>= 1) {
    ps += __shfl_xor(ps, off, 32);
    pd += __shfl_xor(pd, off, 32);
  }
  if (lane == 0) { asrc[i] = ps; adst[i] = pd; }
}

// monotonic float<->uint mapping for atomicMax on signed floats
__device__ __forceinline__ unsigned fkey(float f) {
  unsigned u = __float_as_uint(f);
  return (u & 0x80000000u) ? ~u : (u | 0x80000000u);
}
__device__ __forceinline__ float fkey_inv(unsigned k) {
  return __uint_as_float((k & 0x80000000u) ? (k ^ 0x80000000u) : ~k);
}
__device__ __forceinline__ float leaky(float x) { return x > 0.f ? x : 0.2f * x; }

__global__ void k_gat_max(const int* __restrict__ row, const int* __restrict__ col,
                          const float* __restrict__ asrc, const float* __restrict__ adst,
                          unsigned* __restrict__ emax, int E, int N) {
  int e = blockIdx.x * blockDim.x + threadIdx.x;
  if (e >= E + N) return;
  int r = (e < E) ? row[e] : (e - E);
  int c = (e < E) ? col[e] : (e - E);
  atomicMax(&emax[c], fkey(leaky(asrc[r] + adst[c])));
}

__global__ void k_gat_exp(const int* __restrict__ row, const int* __restrict__ col,
                          const float* __restrict__ asrc, const float* __restrict__ adst,
                          const unsigned* __restrict__ emax, float* __restrict__ esum,
                          float* __restrict__ ebuf, int E, int N) {
  int e = blockIdx.x * blockDim.x + threadIdx.x;
  if (e >= E + N) return;
  int r = (e < E) ? row[e] : (e - E);
  int c = (e < E) ? col[e] : (e - E);
  float ev = __expf(leaky(asrc[r] + adst[c]) - fkey_inv(emax[c]));
  ebuf[e] = ev;
  atomicAdd(&esum[c], ev);
}

// ---------------- weighted feature scatter: dst[col] += w * src[row] ----------------
// MODE 0: w = 1, E edges only (nsum).  MODE 1: w = dinv[r]*dinv[c], E+N (GCN).
// MODE 2: w = ebuf[e], E+N (GAT, unnormalized).  One wave per edge, float4 per lane.
template <int MODE>
__global__ void k_scatter(const float* __restrict__ src, float* __restrict__ dst,
                          const int* __restrict__ row, const int* __restrict__ col,
                          const float* __restrict__ wsrc, int E, int N) {
  long long t = (long long)blockIdx.x * blockDim.x + threadIdx.x;
  long long total = (MODE == 0) ? (long long)E : (long long)(E + N);
  long long e = t >> 5;
  if (e >= total) return;
  int q = (int)(t & 31);
  int r = (e < E) ? row[(int)e] : (int)(e - E);
  int c = (e < E) ? col[(int)e] : (int)(e - E);
  float w = 1.f;
  if (MODE == 1) w = wsrc[r] * wsrc[c];
  if (MODE == 2) w = wsrc[e];
  float4 v = ((const float4*)(src + (size_t)r * DF))[q];
  float* d = dst + (size_t)c * DF + (size_t)q * 4;
  atomicAdd(d + 0, v.x * w);
  atomicAdd(d + 1, v.y * w);
  atomicAdd(d + 2, v.z * w);
  atomicAdd(d + 3, v.w * w);
}

// ---------------- final: relu(acc + gat_acc/esum + all biases) ----------------
__global__ void k_final(const float* __restrict__ acc, const float* __restrict__ gatacc,
                        const float* __restrict__ esum, const float* __restrict__ b1,
                        const float* __restrict__ b2, const float* __restrict__ b3,
                        const float* __restrict__ b4, float* __restrict__ out, int N) {
  long long t = (long long)blockIdx.x * blockDim.x + threadIdx.x;
  if (t >= (long long)N * DF) return;
  int i = (int)(t >> 7), j = (int)(t & 127);
  float val = acc[t] + gatacc[t] / esum[i] + b1[j] + b2[j] + b3[j] + b4[j];
  out[t] = fmaxf(val, 0.f);
}

// ---------------- launch ----------------

extern "C" void kernel_launch(void* const* d_in, const int* in_sizes, int n_in,
                              void* d_out, int out_size, void* d_ws, size_t ws_size,
                              hipStream_t stream) {
  const float* x       = (const float*)d_in[0];
  const int*   eidx    = (const int*)d_in[1];
  const float* gcn_w   = (const float*)d_in[2];
  const float* gcn_b   = (const float*)d_in[3];
  const float* sage_wl = (const float*)d_in[4];
  const float* sage_bl = (const float*)d_in[5];
  const float* sage_wr = (const float*)d_in[6];
  const float* gin_w1  = (const float*)d_in[7];
  const float* gin_b1  = (const float*)d_in[8];
  const float* gin_w2  = (const float*)d_in[9];
  const float* gin_b2  = (const float*)d_in[10];
  const float* gat_w   = (const float*)d_in[11];
  const float* att_s   = (const float*)d_in[12];
  const float* att_d   = (const float*)d_in[13];
  const float* gat_b   = (const float*)d_in[14];

  const int N = in_sizes[0] / DF;
  const int E = in_sizes[1] / 2;
  const int* row = eidx;       // edge_index[0]
  const int* col = eidx + E;   // edge_index[1]

  // workspace layout (floats): ~132 MB total
  float* ws = (float*)d_ws;
  const size_t NF = (size_t)N * DF;
  float* acc    = ws;              // x1 + x2 + x3 accumulator
  float* gatacc = acc + NF;        // unnormalized GAT accumulator
  float* nsum   = gatacc + NF;     // segment_sum(x[row], col): SAGE mean & GIN agg
  float* hbuf   = nsum + NF;       // h_gcn, later reused as GIN hidden layer
  float* hg     = hbuf + NF;       // x @ gat_w
  float* dinv   = hg + NF;
  float* rinv   = dinv + N;
  float* cnt    = rinv + N;
  float* asrc   = cnt + N;
  float* adst   = asrc + N;
  unsigned* emax = (unsigned*)(adst + N);
  float* esum   = adst + 2 * (size_t)N;
  float* ebuf   = esum + N;
  const size_t ztotal = 5 * NF + 7 * (size_t)N + (size_t)(E + N);

  const int B = 256;
  const unsigned gz = (unsigned)((ztotal + (size_t)B - 1) / B);
  k_zero<<<gz, B, 0, stream>>>(ws, ztotal);

  k_cnt<<<(E + B - 1) / B, B, 0, stream>>>(col, cnt, E);
  k_dinv<<<(N + B - 1) / B, B, 0, stream>>>(cnt, dinv, rinv, N);

  const int tiles = (N + 15) / 16;        // 3125 row tiles
  const int gg = (tiles + 7) / 8;         // 8 waves per block
  k_gemm<0, false, false><<<gg, 256, 0, stream>>>(x, nullptr, nullptr, gcn_w, nullptr, hbuf, N);
  k_gemm<0, false, false><<<gg, 256, 0, stream>>>(x, nullptr, nullptr, gat_w, nullptr, hg, N);
  k_att<<<(N + 7) / 8, 256, 0, stream>>>(hg, att_s, att_d, asrc, adst, N);

  const long long sth0 = (long long)E * 32;
  const long long sth1 = (long long)(E + N) * 32;
  k_scatter<0><<<(unsigned)((sth0 + B - 1) / B), B, 0, stream>>>(x, nsum, row, col, nullptr, E, N);
  k_scatter<1><<<(unsigned)((sth1 + B - 1) / B), B, 0, stream>>>(hbuf, acc, row, col, dinv, E, N);

  k_gat_max<<<(E + N + B - 1) / B, B, 0, stream>>>(row, col, asrc, adst, emax, E, N);
  k_gat_exp<<<(E + N + B - 1) / B, B, 0, stream>>>(row, col, asrc, adst, emax, esum, ebuf, E, N);
  k_scatter<2><<<(unsigned)((sth1 + B - 1) / B), B, 0, stream>>>(hg, gatacc, row, col, ebuf, E, N);

  // acc += x @ sage_wr ; acc += mean @ sage_wl ; GIN MLP (hbuf reused) ; acc += mid @ gin_w2
  k_gemm<0, true,  false><<<gg, 256, 0, stream>>>(x,    nullptr, nullptr, sage_wr, nullptr, acc,  N);
  k_gemm<2, true,  false><<<gg, 256, 0, stream>>>(nsum, nullptr, rinv,    sage_wl, nullptr, acc,  N);
  k_gemm<1, false, true ><<<gg, 256, 0, stream>>>(x,    nsum,    nullptr, gin_w1,  gin_b1,  hbuf, N);
  k_gemm<0, true,  false><<<gg, 256, 0, stream>>>(hbuf, nullptr, nullptr, gin_w2,  nullptr, acc,  N);

  k_final<<<(unsigned)((NF + (size_t)B - 1) / B), B, 0, stream>>>(
      acc, gatacc, esum, gcn_b, sage_bl, gin_b2, gat_b, (float*)d_out, N);
}